// ESABotRGAT_32590211842599
// MI455X (gfx1250) — compile-verified
//
#include <hip/hip_runtime.h>

// ---------- CDNA5 WMMA types ----------
typedef __attribute__((ext_vector_type(2))) float v2f;
typedef __attribute__((ext_vector_type(8))) float v8f;

#define KCHUNK 32   // K-chunk staged into LDS per step; all GEMM K's (768,128) divide it

__device__ __forceinline__ float lrelu_f(float x, float s) { return x >= 0.f ? x : s * x; }

// Monotonic order-preserving float<->uint key (atomicMax on floats incl. negatives).
// key(a) < key(b) <=> a < b. Key 0 is below every finite float's key (acts as -inf).
__device__ __forceinline__ unsigned fkey(float f) {
    unsigned u = __float_as_uint(f);
    return (u & 0x80000000u) ? ~u : (u | 0x80000000u);
}
__device__ __forceinline__ float fval(unsigned k) {
    return (k & 0x80000000u) ? __uint_as_float(k ^ 0x80000000u) : __uint_as_float(~k);
}

// ---------------------------------------------------------------------------
// fp32 WMMA GEMM: out[z] = act( A @ W[z] + bias ). One wave per 32x32 output tile:
// 2 row-tiles x 2 col-tiles -> 4 accumulators; each B fragment feeds 2 WMMAs and
// each A fragment feeds 2 WMMAs (4 wmma : 4 W-loads : 2 LDS reads per K-step of 4).
// A rows are staged through LDS with coalesced global_load_async_to_lds_b128
// (8 x 512B per 32x32-float chunk), fragments read back as ds_load_b64.
// A: [M,K] row-major, W: [K,Ncols] row-major, out: [M,ldo] column window.
// blockIdx.z batches over relations via wStrideZ/oStrideZ. K % 32 == 0 required.
// ---------------------------------------------------------------------------
__global__ __launch_bounds__(32)
void rgat_gemm(const float* __restrict__ A, const float* __restrict__ W,
               const float* __restrict__ bias, float* __restrict__ outp,
               int M, int K, int Ncols, int ldo,
               long wStrideZ, long oStrideZ, int act, float slope)
{
    __shared__ float As[32 * KCHUNK];   // 4 KB staging buffer (one wave per block)

    const float* Wz = W + (size_t)blockIdx.z * wStrideZ;
    float* Oz = outp + (size_t)blockIdx.z * oStrideZ;

    const int row0 = blockIdx.x * 32;      // two 16-row tiles per wave
    const int col0 = blockIdx.y * 32;      // two 16-col tiles per wave
    const int lane = threadIdx.x;          // 0..31, one full wave32
    const int lo = lane & 15;
    const int hi = lane >> 4;              // 0 or 1

    // ---- async staging addresses: instr i covers rows i*4 + (lane>>3), 16B per lane ----
    const int srow = lane >> 3;            // 0..3
    const int scol = (lane & 7) * 4;       // 0,4,...,28  (floats)
    unsigned lds_a[8];
    unsigned long long ga[8];
#pragma unroll
    for (int i = 0; i < 8; ++i) {
        const int r = i * 4 + srow;
        int gr = row0 + r; if (gr >= M) gr = M - 1;   // clamp loads; stores row-guarded
        lds_a[i] = (unsigned)(uintptr_t)&As[r * KCHUNK + scol];
        ga[i] = (unsigned long long)(uintptr_t)(A + (size_t)gr * K + scol);
    }

    // ---- B column setup: clamp for loads, 0/1 mask folds OOB contributions ----
    const int c0 = col0 + lo;
    const int c1 = c0 + 16;
    const float m0 = (c0 < Ncols) ? 1.f : 0.f;
    const float m1 = (c1 < Ncols) ? 1.f : 0.f;
    const int c0L = (c0 < Ncols) ? c0 : (Ncols - 1);
    const int c1L = (c1 < Ncols) ? c1 : (Ncols - 1);
    const float* wp0 = Wz + (size_t)(2 * hi) * Ncols + c0L;  // K-row (2*hi), col tile 0
    const float* wp1 = Wz + (size_t)(2 * hi) * Ncols + c1L;  // K-row (2*hi), col tile 1

    v8f acc00 = {0.f, 0.f, 0.f, 0.f, 0.f, 0.f, 0.f, 0.f};
    v8f acc01 = {0.f, 0.f, 0.f, 0.f, 0.f, 0.f, 0.f, 0.f};
    v8f acc10 = {0.f, 0.f, 0.f, 0.f, 0.f, 0.f, 0.f, 0.f};
    v8f acc11 = {0.f, 0.f, 0.f, 0.f, 0.f, 0.f, 0.f, 0.f};

    for (int k0 = 0; k0 < K; k0 += KCHUNK) {
        // Stage 32 x KCHUNK A tile into LDS: 8 coalesced 512B async transfers.
#pragma unroll
        for (int i = 0; i < 8; ++i) {
            asm volatile("global_load_async_to_lds_b128 %0, %1, off"
                         :: "v"(lds_a[i]), "v"(ga[i]) : "memory");
            ga[i] += KCHUNK * 4;
        }
        asm volatile("s_wait_asynccnt 0x0" ::: "memory");

#pragma unroll
        for (int kk = 0; kk < KCHUNK; kk += 4) {
            // A 16x4 f32 fragments from LDS: lanes 0-15 -> K0/K1, 16-31 -> K2/K3
            const float2 a0f = *(const float2*)&As[lo * KCHUNK + kk + 2 * hi];
            const float2 a1f = *(const float2*)&As[(lo + 16) * KCHUNK + kk + 2 * hi];
            v2f a0; a0.x = a0f.x; a0.y = a0f.y;
            v2f a1; a1.x = a1f.x; a1.y = a1f.y;
            // B 4x16 fragments: VGPR0 = K-rows {0,2}, VGPR1 = K-rows {1,3}
            v2f b0; b0.x = wp0[0] * m0; b0.y = wp0[Ncols] * m0;
            v2f b1; b1.x = wp1[0] * m1; b1.y = wp1[Ncols] * m1;
            acc00 = __builtin_amdgcn_wmma_f32_16x16x4_f32(false, a0, false, b0,
                                                          (short)0, acc00, false, false);
            acc01 = __builtin_amdgcn_wmma_f32_16x16x4_f32(false, a0, false, b1,
                                                          (short)0, acc01, false, false);
            acc10 = __builtin_amdgcn_wmma_f32_16x16x4_f32(false, a1, false, b0,
                                                          (short)0, acc10, false, false);
            acc11 = __builtin_amdgcn_wmma_f32_16x16x4_f32(false, a1, false, b1,
                                                          (short)0, acc11, false, false);
            wp0 += (size_t)4 * Ncols;
            wp1 += (size_t)4 * Ncols;
        }
    }

    // ---- epilogue: C/D layout VGPR r = rows {r, r+8}; cols = lanes ----
    const bool s0 = (c0 < Ncols), s1 = (c1 < Ncols);
    const float bv0 = (bias && s0) ? bias[c0] : 0.f;
    const float bv1 = (bias && s1) ? bias[c1] : 0.f;
#pragma unroll
    for (int r = 0; r < 8; ++r) {
        const int rowA = row0 + r + 8 * hi;
        const int rowB = rowA + 16;
        if (rowA < M) {
            if (s0) { float v = acc00[r] + bv0; if (act) v = lrelu_f(v, slope);
                      Oz[(size_t)rowA * ldo + c0] = v; }
            if (s1) { float v = acc01[r] + bv1; if (act) v = lrelu_f(v, slope);
                      Oz[(size_t)rowA * ldo + c1] = v; }
        }
        if (rowB < M) {
            if (s0) { float v = acc10[r] + bv0; if (act) v = lrelu_f(v, slope);
                      Oz[(size_t)rowB * ldo + c0] = v; }
            if (s1) { float v = acc11[r] + bv1; if (act) v = lrelu_f(v, slope);
                      Oz[(size_t)rowB * ldo + c1] = v; }
        }
    }
}

// ---------------------------------------------------------------------------
// Small feature encoders (K = 7 / 11 / 1): scalar, writes x0 cols 64..127.
// Concat layout: des[0:28) tweet[28:64) num[64:76) cat[76:116) newf[116:128)
// ---------------------------------------------------------------------------
__global__ __launch_bounds__(256)
void rgat_small_enc(const float* __restrict__ num_prop, const float* __restrict__ cat_prop,
                    const float* __restrict__ newf,
                    const float* __restrict__ Wn, const float* __restrict__ bn,
                    const float* __restrict__ Wc, const float* __restrict__ bc,
                    const float* __restrict__ Wf, const float* __restrict__ bf,
                    float* __restrict__ x0, int n_nodes)
{
    const int idx = blockIdx.x * 256 + threadIdx.x;   // n*64 + c
    if (idx >= n_nodes * 64) return;
    const int n = idx >> 6;
    const int c = idx & 63;
    float acc; int outc;
    if (c < 12) {                       // num_prop [7] -> 12
        acc = bn[c];
        for (int i = 0; i < 7; ++i) acc += num_prop[(size_t)n * 7 + i] * Wn[i * 12 + c];
        outc = 64 + c;
    } else if (c < 52) {                // cat_prop [11] -> 40
        const int cc = c - 12;
        acc = bc[cc];
        for (int i = 0; i < 11; ++i) acc += cat_prop[(size_t)n * 11 + i] * Wc[i * 40 + cc];
        outc = 76 + cc;
    } else {                            // new_feature [1] -> 12
        const int cc = c - 52;
        acc = bf[cc] + newf[n] * Wf[cc];
        outc = 116 + cc;
    }
    x0[(size_t)n * 128 + outc] = lrelu_f(acc, 0.01f);
}

// ---------------------------------------------------------------------------
// Per-node attention projections: qi[r,n,h]=xw[r,n]·q[:,h], kj likewise.
// Factors the reference's per-edge 128-dots into per-node work.
// ---------------------------------------------------------------------------
template <int H>
__global__ __launch_bounds__(256)
void rgat_qk(const float* __restrict__ xw, const float* __restrict__ q,
             const float* __restrict__ k, float* __restrict__ qi,
             float* __restrict__ kj, int n_nodes)
{
    const int idx = blockIdx.x * 256 + threadIdx.x;   // r*N + n
    if (idx >= 2 * n_nodes) return;
    const float* xr = xw + (size_t)idx * 128;
    float qa[H], ka[H];
#pragma unroll
    for (int h = 0; h < H; ++h) { qa[h] = 0.f; ka[h] = 0.f; }
    for (int c = 0; c < 128; ++c) {
        const float xv = xr[c];
#pragma unroll
        for (int h = 0; h < H; ++h) {
            qa[h] += xv * q[c * H + h];
            ka[h] += xv * k[c * H + h];
        }
    }
#pragma unroll
    for (int h = 0; h < H; ++h) {
        qi[(size_t)idx * H + h] = qa[h];
        kj[(size_t)idx * H + h] = ka[h];
    }
}

// Per-layer init: agg = bias (broadcast), denom = 0, amax key = 0 (== -inf sentinel).
__global__ __launch_bounds__(256)
void rgat_init(float* __restrict__ agg, float* __restrict__ denom,
               unsigned* __restrict__ amaxKey, const float* __restrict__ bias,
               int n_nodes, int H)
{
    const int idx = blockIdx.x * 256 + threadIdx.x;   // n*128 + c
    if (idx >= n_nodes * 128) return;
    const int c = idx & 127;
    agg[idx] = bias[c];
    if (c < H) {
        const int n = idx >> 7;
        denom[(size_t)n * H + c] = 0.f;
        amaxKey[(size_t)n * H + c] = 0u;
    }
}

// Pass 1: alpha[e,h] = lrelu(qi[t,dst,h] + kj[t,src,h], 0.2); segment max via key atomicMax.
template <int H>
__global__ __launch_bounds__(256)
void rgat_alpha(const int* __restrict__ ei, const int* __restrict__ et,
                const float* __restrict__ qi, const float* __restrict__ kj,
                float* __restrict__ alpha, unsigned* __restrict__ amaxKey,
                int n_edges, int n_nodes)
{
    const int e = blockIdx.x * 256 + threadIdx.x;
    if (e >= n_edges) return;
    const int src = ei[e];
    const int dst = ei[n_edges + e];
    const int t = et[e];
    const size_t qb = ((size_t)t * n_nodes + dst) * H;
    const size_t kb = ((size_t)t * n_nodes + src) * H;
#pragma unroll
    for (int h = 0; h < H; ++h) {
        const float a = lrelu_f(qi[qb + h] + kj[kb + h], 0.2f);
        alpha[(size_t)e * H + h] = a;
        atomicMax(&amaxKey[(size_t)dst * H + h], fkey(a));
    }
}

// Pass 2: ex = exp(alpha - amax[dst]) (in place); denom[dst] += ex.
template <int H>
__global__ __launch_bounds__(256)
void rgat_ex(const int* __restrict__ ei, float* __restrict__ alpha,
             const unsigned* __restrict__ amaxKey, float* __restrict__ denom,
             int n_edges)
{
    const int idx = blockIdx.x * 256 + threadIdx.x;   // e*H + h
    if (idx >= n_edges * H) return;
    const int e = idx / H;
    const int h = idx - e * H;
    const int dst = ei[n_edges + e];
    const float am = fval(amaxKey[(size_t)dst * H + h]);
    const float v = expf(alpha[idx] - am);
    alpha[idx] = v;
    atomicAdd(&denom[(size_t)dst * H + h], v);
}

// Pass 3: one wave per edge; lane handles 4 channels (float4 gather, 4 f32 atomics).
// agg[dst,c] += (ex/denom) * xw[t,src,c].  xw/agg are L2-resident on MI455X (51+25 MB << 192 MB).
template <int H>
__global__ __launch_bounds__(256)
void rgat_scatter(const int* __restrict__ ei, const int* __restrict__ et,
                  const float* __restrict__ ex, const float* __restrict__ denom,
                  const float* __restrict__ xw, float* __restrict__ agg,
                  int n_edges, int n_nodes)
{
    const int gid = blockIdx.x * 256 + threadIdx.x;
    const int e = gid >> 5;
    if (e >= n_edges) return;
    const int lane = gid & 31;
    const int c0 = lane * 4;
    const int src = ei[e];
    const int dst = ei[n_edges + e];
    const int t = et[e];
    constexpr int C = 128 / H;
    const int h = c0 / C;
    const float attn = ex[(size_t)e * H + h] /
                       (denom[(size_t)dst * H + h] + 1e-16f);
    const float4 xv = *(const float4*)(xw + (((size_t)t * n_nodes + src) << 7) + c0);
    float* ag = agg + (((size_t)dst) << 7) + c0;
    atomicAdd(ag + 0, attn * xv.x);
    atomicAdd(ag + 1, attn * xv.y);
    atomicAdd(ag + 2, attn * xv.z);
    atomicAdd(ag + 3, attn * xv.w);
}

// ---------------------------------------------------------------------------
extern "C" void kernel_launch(void* const* d_in, const int* in_sizes, int n_in,
                              void* d_out, int out_size, void* d_ws, size_t ws_size,
                              hipStream_t stream)
{
    const float* des  = (const float*)d_in[0];
    const float* twt  = (const float*)d_in[1];
    const float* nump = (const float*)d_in[2];
    const float* catp = (const float*)d_in[3];
    const float* newf = (const float*)d_in[4];
    const int* edge_index = (const int*)d_in[5];
    const int* edge_type  = (const int*)d_in[6];
    const float *Wd = (const float*)d_in[7],  *bd = (const float*)d_in[8];
    const float *Wt = (const float*)d_in[9],  *bt = (const float*)d_in[10];
    const float *Wn = (const float*)d_in[11], *bn = (const float*)d_in[12];
    const float *Wc = (const float*)d_in[13], *bc = (const float*)d_in[14];
    const float *Wf = (const float*)d_in[15], *bf = (const float*)d_in[16];
    const float *Wi = (const float*)d_in[17], *bi = (const float*)d_in[18];
    const float *w1 = (const float*)d_in[19], *q1 = (const float*)d_in[20];
    const float *k1 = (const float*)d_in[21], *b1 = (const float*)d_in[22];
    const float *w2 = (const float*)d_in[23], *q2 = (const float*)d_in[24];
    const float *k2 = (const float*)d_in[25], *b2 = (const float*)d_in[26];
    const float *Wo1 = (const float*)d_in[27], *bo1 = (const float*)d_in[28];
    const float *Wo2 = (const float*)d_in[29], *bo2 = (const float*)d_in[30];
    float* out = (float*)d_out;

    const int N = in_sizes[0] / 768;   // 50000
    const int E = in_sizes[6];         // 800000

    // Workspace carve-out (floats). x1 is reused as agg2, x0 as x3 after they expire.
    float* ws = (float*)d_ws;
    size_t off = 0;
    auto carve = [&](size_t n) { float* p = ws + off; off += n; return p; };
    float* x0    = carve((size_t)N * 128);
    float* x1    = carve((size_t)N * 128);
    float* xw    = carve((size_t)2 * N * 128);
    float* agg1  = carve((size_t)N * 128);
    float* qi    = carve((size_t)2 * N * 4);
    float* kj    = carve((size_t)2 * N * 4);
    float* alpha = carve((size_t)E * 4);
    float* denom = carve((size_t)N * 4);
    unsigned* amaxKey = (unsigned*)carve((size_t)N * 4);
    float* agg2 = x1;   // x1 dead after layer-1's xw GEMM
    float* x3   = x0;   // x0 dead after input linear

    const int rt = (N + 31) / 32;      // 1563 row tiles (32 rows per wave)
    const dim3 blk32(32, 1, 1), blk256(256, 1, 1);

    // --- feature encoders -> x0[N,128] ---   (grid.y = ceil(Ncols/32))
    rgat_gemm<<<dim3(rt, 1, 1), blk32, 0, stream>>>(des, Wd, bd, x0 + 0, N, 768, 28, 128, 0, 0, 1, 0.01f);
    rgat_gemm<<<dim3(rt, 2, 1), blk32, 0, stream>>>(twt, Wt, bt, x0 + 28, N, 768, 36, 128, 0, 0, 1, 0.01f);
    rgat_small_enc<<<(N * 64 + 255) / 256, blk256, 0, stream>>>(nump, catp, newf, Wn, bn, Wc, bc, Wf, bf, x0, N);

    // --- input linear: x1 = lrelu(x0 @ Wi + bi) ---
    rgat_gemm<<<dim3(rt, 4, 1), blk32, 0, stream>>>(x0, Wi, bi, x1, N, 128, 128, 128, 0, 0, 1, 0.01f);

    // --- RGAT layer 1 (H=4, C=32): x1 -> agg1 ---
    rgat_gemm<<<dim3(rt, 4, 2), blk32, 0, stream>>>(x1, w1, nullptr, xw, N, 128, 128, 128,
                                                    (long)128 * 128, (long)N * 128, 0, 0.f);
    rgat_qk<4><<<(2 * N + 255) / 256, blk256, 0, stream>>>(xw, q1, k1, qi, kj, N);
    rgat_init<<<(N * 128 + 255) / 256, blk256, 0, stream>>>(agg1, denom, amaxKey, b1, N, 4);
    rgat_alpha<4><<<(E + 255) / 256, blk256, 0, stream>>>(edge_index, edge_type, qi, kj, alpha, amaxKey, E, N);
    rgat_ex<4><<<(E * 4 + 255) / 256, blk256, 0, stream>>>(edge_index, alpha, amaxKey, denom, E);
    rgat_scatter<4><<<(E * 32 + 255) / 256, blk256, 0, stream>>>(edge_index, edge_type, alpha, denom, xw, agg1, E, N);

    // --- RGAT layer 2 (H=1, C=128): agg1 -> agg2 ---
    rgat_gemm<<<dim3(rt, 4, 2), blk32, 0, stream>>>(agg1, w2, nullptr, xw, N, 128, 128, 128,
                                                    (long)128 * 128, (long)N * 128, 0, 0.f);
    rgat_qk<1><<<(2 * N + 255) / 256, blk256, 0, stream>>>(xw, q2, k2, qi, kj, N);
    rgat_init<<<(N * 128 + 255) / 256, blk256, 0, stream>>>(agg2, denom, amaxKey, b2, N, 1);
    rgat_alpha<1><<<(E + 255) / 256, blk256, 0, stream>>>(edge_index, edge_type, qi, kj, alpha, amaxKey, E, N);
    rgat_ex<1><<<(E + 255) / 256, blk256, 0, stream>>>(edge_index, alpha, amaxKey, denom, E);
    rgat_scatter<1><<<(E * 32 + 255) / 256, blk256, 0, stream>>>(edge_index, edge_type, alpha, denom, xw, agg2, E, N);

    // --- output MLP ---
    rgat_gemm<<<dim3(rt, 4, 1), blk32, 0, stream>>>(agg2, Wo1, bo1, x3, N, 128, 128, 128, 0, 0, 1, 0.01f);
    rgat_gemm<<<dim3(rt, 1, 1), blk32, 0, stream>>>(x3, Wo2, bo2, out, N, 128, 2, 2, 0, 0, 0, 0.f);
}